// OneFlashAttention_36721970381392
// MI455X (gfx1250) — compile-verified
//
#include <hip/hip_runtime.h>
#include <math.h>

// fp32 WMMA vector types: A/B for 16x16x4 f32 = 2 VGPRs, C/D = 8 VGPRs
typedef __attribute__((ext_vector_type(2))) float v2f;
typedef __attribute__((ext_vector_type(8))) float v8f;

// 16-byte int vector type matching the async-to-LDS builtin's pointee type
typedef int v4i_b __attribute__((vector_size(4 * sizeof(int))));

// Problem shape (fixed by the reference)
constexpr int HD   = 128;   // head dim
constexpr int BLK  = 64;    // block size
constexpr int SEQ  = 4096;
constexpr int NH   = 16;
constexpr int NBAT = 2;
// Separate padded LDS row strides, each conflict-free for its access phase:
//  K is read row-across-lanes  -> need stride ≡ 4 (mod 64)  -> 132
//  V is read col-across-lanes with half-waves 2 rows apart -> need stride ≡ 8 (mod 64) -> 136
constexpr int KSTR = 132;
constexpr int VSTR = 136;

// gfx1250 async global->LDS path (ASYNCcnt), with sync fallback
#define ASYNC_OK (__has_builtin(__builtin_amdgcn_global_load_async_to_lds_b128) && \
                  __has_builtin(__builtin_amdgcn_s_wait_asynccnt))

#if ASYNC_OK
#define WAIT_ASYNC(n) __builtin_amdgcn_s_wait_asynccnt(n)
#define GAS1(p) ((__attribute__((address_space(1))) v4i_b*)(p))
#define LAS3(p) ((__attribute__((address_space(3))) v4i_b*)(p))
#else
#define WAIT_ASYNC(n) ((void)0)
#endif

__device__ __forceinline__ float clip1e4(float x) {
    return fminf(fmaxf(x, -10000.0f), 10000.0f);
}

// Stage one 64x128 f32 block (coalesced b128) into LDS with padded rows.
// 128 threads x 16 transfers. Async path increments ASYNCcnt by 16 per wave.
template <int STR>
__device__ __forceinline__ void stage_block(const float* __restrict__ g, float* l, int t) {
    #pragma unroll
    for (int j = 0; j < 16; ++j) {
        const int i   = t + 128 * j;     // float4 index within the 64x128 block
        const int row = i >> 5;          // 32 float4 per row
        const int col = (i & 31) << 2;   // float column
#if ASYNC_OK
        __builtin_amdgcn_global_load_async_to_lds_b128(
            GAS1(g + (size_t)i * 4),
            LAS3(l + row * STR + col),
            0, 0);
#else
        *(float4*)(l + row * STR + col) = *(const float4*)(g + (size_t)i * 4);
#endif
    }
}

__launch_bounds__(128)
__global__ void blockwise_attn_kernel(const float* __restrict__ q,
                                      const float* __restrict__ k,
                                      const float* __restrict__ v,
                                      float* __restrict__ out)
{
    __shared__ float kls[2][BLK * KSTR];   // double-buffered K blocks (stride 132)
    __shared__ float vls[2][BLK * VSTR];   // double-buffered V blocks (stride 136)
    __shared__ float sc [4][16][BLK];      // per-wave E tile (D-layout -> A-layout change)

    const int mblk = blockIdx.x;           // query block 0..63
    const int h    = blockIdx.y;
    const int b    = blockIdx.z;
    const int t    = threadIdx.x;
    const int wave = t >> 5;               // 0..3 (wave32)
    const int lane = t & 31;
    const int lh   = lane & 15;            // 0..15
    const int hi   = lane >> 4;            // half-wave select

    const size_t baseBH = ((size_t)b * NH + h) * (size_t)SEQ * HD;
    const int    m0     = mblk * BLK + wave * 16;
    const float  scale  = 0.08838834764831845f;     // 1/sqrt(128)

    const int nIter = (mblk == 0) ? 1 : 2;          // prev block (if any), then diagonal
    const int nblk0 = (mblk == 0) ? 0 : (mblk - 1);

    // ---- issue ALL staging up-front (async): K0, V0, (K1, V1) -------------
    stage_block<KSTR>(k + baseBH + (size_t)nblk0 * BLK * HD, kls[0], t);
    stage_block<VSTR>(v + baseBH + (size_t)nblk0 * BLK * HD, vls[0], t);
    if (nIter == 2) {
        stage_block<KSTR>(k + baseBH + (size_t)mblk * BLK * HD, kls[1], t);
        stage_block<VSTR>(v + baseBH + (size_t)mblk * BLK * HD, vls[1], t);
    }

    // ---- Q slab into registers in WMMA A-layout (overlaps async staging) --
    // A 16x4 f32: lane holds row M=lh, VGPR pair = K = 4s + 2*hi + {0,1}
    v2f qa[32];
    {
        const float* qrow = q + baseBH + (size_t)(m0 + lh) * HD;
        #pragma unroll
        for (int s = 0; s < 32; ++s) {
            const int kk = 4 * s + 2 * hi;
            v2f tq;
            tq.x = clip1e4(qrow[kk])     * scale;
            tq.y = clip1e4(qrow[kk + 1]) * scale;
            qa[s] = tq;
        }
    }

    v8f acc_o[8];
    #pragma unroll
    for (int d = 0; d < 8; ++d) acc_o[d] = {};
    float lsum_tot[8];
    #pragma unroll
    for (int r = 0; r < 8; ++r) lsum_tot[r] = 0.0f;

    for (int it = 0; it < nIter; ++it) {
        const bool   diag = (it == nIter - 1);
        const float* kl   = kls[it];
        const float* vl   = vls[it];

        // need this iter's K tile (staged waits keep later loads in flight)
        if (nIter == 2 && it == 0) WAIT_ASYNC(48);
        else                       WAIT_ASYNC(16);
        __syncthreads();

        // ---- S(16x64) = Qslab @ Kblk^T : 4 N-tiles, K from LDS ------------
        v8f acc[4];
        #pragma unroll
        for (int nt = 0; nt < 4; ++nt) acc[nt] = {};
        #pragma unroll
        for (int s = 0; s < 32; ++s) {
            const int kk = 4 * s + 2 * hi;
            #pragma unroll
            for (int nt = 0; nt < 4; ++nt) {
                v2f bv = *(const v2f*)&kl[(nt * 16 + lh) * KSTR + kk];
                bv.x = clip1e4(bv.x);
                bv.y = clip1e4(bv.y);
                acc[nt] = __builtin_amdgcn_wmma_f32_16x16x4_f32(
                            false, qa[s], false, bv, (short)0, acc[nt], false, false);
            }
        }

        // ---- tril mask inside the diagonal block --------------------------
        if (diag) {
            #pragma unroll
            for (int nt = 0; nt < 4; ++nt) {
                const int jloc = nt * 16 + lh;
                #pragma unroll
                for (int r = 0; r < 8; ++r)
                    if (jloc > wave * 16 + r + 8 * hi) acc[nt][r] = -INFINITY;
            }
        }

        // ---- row max / exp(+1e-10) / row sum, all in registers ------------
        // D-layout: each half-wave holds complete rows -> reduce over nt then 16 lanes
        float m[8];
        #pragma unroll
        for (int r = 0; r < 8; ++r) {
            float mm = fmaxf(fmaxf(acc[0][r], acc[1][r]), fmaxf(acc[2][r], acc[3][r]));
            mm = fmaxf(mm, __shfl_xor(mm, 1, 32));
            mm = fmaxf(mm, __shfl_xor(mm, 2, 32));
            mm = fmaxf(mm, __shfl_xor(mm, 4, 32));
            mm = fmaxf(mm, __shfl_xor(mm, 8, 32));
            m[r] = mm;
        }
        #pragma unroll
        for (int r = 0; r < 8; ++r) {
            float s0 = 0.0f;
            #pragma unroll
            for (int nt = 0; nt < 4; ++nt) {
                const float e = __expf(acc[nt][r] - m[r]) + 1e-10f;
                sc[wave][r + 8 * hi][nt * 16 + lh] = e;   // E in D-layout -> LDS
                s0 += e;
            }
            s0 += __shfl_xor(s0, 1, 32);
            s0 += __shfl_xor(s0, 2, 32);
            s0 += __shfl_xor(s0, 4, 32);
            s0 += __shfl_xor(s0, 8, 32);
            lsum_tot[r] += s0;
        }

        // need this iter's V tile
        if (diag) WAIT_ASYNC(0);
        else      WAIT_ASYNC(32);
        __syncthreads();

        // ---- O(16x128) += E(16x64) @ Vblk : E and V from LDS --------------
        #pragma unroll
        for (int s2 = 0; s2 < 16; ++s2) {
            const int kk = 4 * s2 + 2 * hi;
            const v2f av = *(const v2f*)&sc[wave][lh][kk];   // E in A-layout
            #pragma unroll
            for (int dt = 0; dt < 8; ++dt) {
                v2f bv;
                bv.x = vl[kk       * VSTR + dt * 16 + lh];
                bv.y = vl[(kk + 1) * VSTR + dt * 16 + lh];
                acc_o[dt] = __builtin_amdgcn_wmma_f32_16x16x4_f32(
                              false, av, false, bv, (short)0, acc_o[dt], false, false);
            }
        }
    }

    // ---- out = o / max(l, 1e-10), nan_to_num; l already lives per-lane ----
    #pragma unroll
    for (int dt = 0; dt < 8; ++dt) {
        #pragma unroll
        for (int r = 0; r < 8; ++r) {
            const int   M     = r + 8 * hi;
            const float denom = fmaxf(lsum_tot[r], 1e-10f);
            float val = acc_o[dt][r] / denom;
            if (!isfinite(val)) val = 0.0f;
            out[baseBH + (size_t)(m0 + M) * HD + dt * 16 + lh] = val;
        }
    }
}

extern "C" void kernel_launch(void* const* d_in, const int* in_sizes, int n_in,
                              void* d_out, int out_size, void* d_ws, size_t ws_size,
                              hipStream_t stream) {
    (void)in_sizes; (void)n_in; (void)out_size; (void)d_ws; (void)ws_size;
    const float* q = (const float*)d_in[0];
    const float* k = (const float*)d_in[1];
    const float* v = (const float*)d_in[2];
    float* out = (float*)d_out;

    dim3 grid(SEQ / BLK, NH, NBAT);   // 64 x 16 x 2 = 2048 workgroups
    dim3 block(128);                  // 4 waves (wave32)
    blockwise_attn_kernel<<<grid, block, 0, stream>>>(q, k, v, out);
}